// GSAGEWrapper_34041910788824
// MI455X (gfx1250) — compile-verified
//
#include <hip/hip_runtime.h>

#define N_NODES 50000
#define N_EDGES 800000
#define HID 128
#define PRED_LEN 24
#define N_LAYERS 3

typedef __attribute__((ext_vector_type(2))) float v2f;
typedef __attribute__((ext_vector_type(8))) float v8f;

// ---------------------------------------------------------------------------
// Kernel 1: edge scatter. One wave (32 lanes) per edge; each lane atomically
// adds 4 coalesced floats of x[src] into agg[dst]; lane 0 bumps the count.
// L2-resident destination (25.6 MB) -> atomics resolve at L2.
// ---------------------------------------------------------------------------
__global__ void sage_scatter(const float* __restrict__ x,
                             const int* __restrict__ src,
                             const int* __restrict__ dst,
                             float* __restrict__ agg,
                             float* __restrict__ cnt) {
  int wave = (int)((blockIdx.x * blockDim.x + threadIdx.x) >> 5);
  int lane = threadIdx.x & 31;
  if (wave >= N_EDGES) return;
  int s = src[wave];
  int d = dst[wave];
  const float* xs = x + (size_t)s * HID;
  float* ad = agg + (size_t)d * HID;
#pragma unroll
  for (int i = 0; i < 4; ++i) {
    int h = i * 32 + lane;  // coalesced within the wave
    atomicAdd(&ad[h], xs[h]);
  }
  if (lane == 0) atomicAdd(&cnt[d], 1.0f);
}

// ---------------------------------------------------------------------------
// Kernel 2: fused SAGE layer GEMM using fp32 WMMA (16x16x4).
//   out = relu( (agg / max(cnt,1)) @ Wl^T + bl + x @ Wr^T )
// Block = 256 threads = 8 waves; block handles one 16-row M tile, wave w
// handles the 16-column N tile w (HID/16 == 8). 32 K-steps x 2 WMMA per wave.
// EXEC is all-ones throughout (no divergence).
// ---------------------------------------------------------------------------
__global__ void sage_gemm(const float* __restrict__ agg,
                          const float* __restrict__ cnt,
                          const float* __restrict__ xin,
                          const float* __restrict__ Wl,   // [HID, HID] (row = out feature)
                          const float* __restrict__ bl,   // [HID]
                          const float* __restrict__ Wr,   // [HID, HID]
                          float* __restrict__ xout) {
  int tileM = blockIdx.x;           // 0..3124
  int wid   = threadIdx.x >> 5;     // 0..7 -> N tile
  int lane  = threadIdx.x & 31;
  int half  = lane >> 4;            // 0: K0/K1, 1: K2/K3
  int l15   = lane & 15;
  int row   = tileM * 16 + l15;     // A-matrix row for this lane
  int col   = wid * 16 + l15;       // B-matrix column for this lane

  float inv = 1.0f / fmaxf(cnt[row], 1.0f);   // mean normalization, folded into A

  const float* aR  = agg + (size_t)row * HID;
  const float* xR  = xin + (size_t)row * HID;
  const float* wlC = Wl + (size_t)col * HID;  // row of Wl == column of Wl^T
  const float* wrC = Wr + (size_t)col * HID;
  int koff = half * 2;

  v8f acc = {};
#pragma unroll 4
  for (int k0 = 0; k0 < HID; k0 += 4) {
    int kk = k0 + koff;
    v2f a1 = *(const v2f*)(aR + kk);
    a1 *= inv;
    v2f b1 = *(const v2f*)(wlC + kk);
    acc = __builtin_amdgcn_wmma_f32_16x16x4_f32(false, a1, false, b1,
                                                (short)0, acc, false, false);
    v2f a2 = *(const v2f*)(xR + kk);
    v2f b2 = *(const v2f*)(wrC + kk);
    acc = __builtin_amdgcn_wmma_f32_16x16x4_f32(false, a2, false, b2,
                                                (short)0, acc, false, false);
  }

  float bias = bl[col];
  // C/D layout: VGPR v, lanes 0-15 -> M=v, lanes 16-31 -> M=v+8; N = lane&15
  float* outBase = xout + ((size_t)(tileM * 16 + half * 8)) * HID + col;
#pragma unroll
  for (int v = 0; v < 8; ++v) {
    float r = acc[v] + bias;
    outBase[(size_t)v * HID] = fmaxf(r, 0.0f);   // ReLU
  }
}

// ---------------------------------------------------------------------------
// Kernel 3: final projection out[N,24] = x @ Wout^T + bout via fp32 WMMA.
// Two 16-column tiles; columns >= 24 are masked to zero in the B operand
// (branchless, EXEC stays all-ones across WMMA) and simply not stored.
// ---------------------------------------------------------------------------
__global__ void sage_proj(const float* __restrict__ xin,
                          const float* __restrict__ Wout,  // [PRED_LEN, HID]
                          const float* __restrict__ bout,  // [PRED_LEN]
                          float* __restrict__ out) {       // [N, PRED_LEN]
  int tileM = blockIdx.x;           // 0..3124
  int wid   = threadIdx.x >> 5;     // 0..1
  int lane  = threadIdx.x & 31;
  int half  = lane >> 4;
  int l15   = lane & 15;
  int row   = tileM * 16 + l15;
  int col   = wid * 16 + l15;

  bool colOK = (col < PRED_LEN);
  float mask = colOK ? 1.0f : 0.0f;
  const float* xR = xin + (size_t)row * HID;
  const float* wC = Wout + (size_t)(colOK ? col : 0) * HID;
  int koff = half * 2;

  v8f acc = {};
#pragma unroll 4
  for (int k0 = 0; k0 < HID; k0 += 4) {
    int kk = k0 + koff;
    v2f a = *(const v2f*)(xR + kk);
    v2f b = *(const v2f*)(wC + kk);
    b *= mask;  // zero out padded columns, no EXEC change
    acc = __builtin_amdgcn_wmma_f32_16x16x4_f32(false, a, false, b,
                                                (short)0, acc, false, false);
  }

  if (colOK) {
    float bias = bout[col];
#pragma unroll
    for (int v = 0; v < 8; ++v) {
      int m = tileM * 16 + half * 8 + v;
      out[(size_t)m * PRED_LEN + col] = acc[v] + bias;
    }
  }
}

// ---------------------------------------------------------------------------
extern "C" void kernel_launch(void* const* d_in, const int* in_sizes, int n_in,
                              void* d_out, int out_size, void* d_ws, size_t ws_size,
                              hipStream_t stream) {
  (void)in_sizes; (void)n_in; (void)out_size; (void)ws_size;
  const float* x    = (const float*)d_in[0];
  const int*   ei   = (const int*)d_in[1];     // [2, E]
  const float* Wl   = (const float*)d_in[2];   // [3, H, H]
  const float* bl   = (const float*)d_in[3];   // [3, H]
  const float* Wr   = (const float*)d_in[4];   // [3, H, H]
  const float* Wout = (const float*)d_in[5];   // [24, H]
  const float* bout = (const float*)d_in[6];   // [24]
  float* out = (float*)d_out;

  const size_t NH = (size_t)N_NODES * HID;
  float* buf0 = (float*)d_ws;       // ping
  float* buf1 = buf0 + NH;          // pong
  float* agg  = buf1 + NH;          // aggregation accumulator
  float* cnt  = agg + NH;           // per-node degree (float)

  const int* src = ei;
  const int* dst = ei + N_EDGES;

  const float* cur = x;
  float* bufs[2] = {buf0, buf1};
  for (int layer = 0; layer < N_LAYERS; ++layer) {
    hipMemsetAsync(agg, 0, NH * sizeof(float), stream);
    hipMemsetAsync(cnt, 0, (size_t)N_NODES * sizeof(float), stream);
    // 8 waves per 256-thread block, one wave per edge: exactly E/8 blocks
    sage_scatter<<<N_EDGES / 8, 256, 0, stream>>>(cur, src, dst, agg, cnt);
    float* nxt = bufs[layer & 1];
    sage_gemm<<<N_NODES / 16, 256, 0, stream>>>(
        agg, cnt, cur,
        Wl + (size_t)layer * HID * HID,
        bl + (size_t)layer * HID,
        Wr + (size_t)layer * HID * HID,
        nxt);
    cur = nxt;
  }
  sage_proj<<<N_NODES / 16, 64, 0, stream>>>(cur, Wout, bout, out);
}